// GResConv_4011499454859
// MI455X (gfx1250) — compile-verified
//
#include <hip/hip_runtime.h>

#define N_NODES 100000
#define N_EDGES 1600000
#define IN_DIM  128
#define OUT_DIM 64
#define N_TILES (N_NODES / 16)   // 6250, exact
#define SP      288              // padded LDS row stride (floats); 288 % 64 == 32 -> bank-conflict-free

typedef __attribute__((ext_vector_type(2))) float v2f;
typedef __attribute__((ext_vector_type(8))) float v8f;

// ---------------------------------------------------------------- utilities
__global__ void zero_kernel(float* __restrict__ p, int n) {
    int i = blockIdx.x * blockDim.x + threadIdx.x;
    if (i < n) p[i] = 0.0f;
}

// ------------------------------------------------------------- degree count
__global__ void degree_kernel(const int* __restrict__ src, const int* __restrict__ dst,
                              float* __restrict__ out_deg, float* __restrict__ in_deg,
                              int nE) {
    int e = blockIdx.x * blockDim.x + threadIdx.x;
    if (e < nE) {
        atomicAdd(&out_deg[src[e]], 1.0f);
        atomicAdd(&in_deg[dst[e]], 1.0f);
    }
}

// ------------------------------------------------------------------- norms
__global__ void norm_kernel(const float* __restrict__ in_deg, const float* __restrict__ out_deg,
                            float* __restrict__ in_norm, float* __restrict__ out_norm, int n) {
    int i = blockIdx.x * blockDim.x + threadIdx.x;
    if (i < n) {
        in_norm[i]  = 1.0f / sqrtf(fmaxf(in_deg[i],  1.0f));
        out_norm[i] = 1.0f / sqrtf(fmaxf(out_deg[i], 1.0f));
    }
}

// -------------------------------------------------- fused dual-GEMM (WMMA)
// z[v] = in_norm[v]*(prev[v]@W_res) + out_norm[v]*(prev[v]@W_conv)
// Each wave computes TWO 16-row tiles, K=128 in 32 steps of
// V_WMMA_F32_16X16X4_F32; each b64 B fragment feeds 2 WMMAs.
// B (4x16) VGPR layout: pair = rows {k+2h, k+2h+1} at column (16t+m), packed
// adjacently in LDS so one ds_load_b64 is the finished operand.
__global__ __launch_bounds__(256)
void gemm_z_kernel(const float* __restrict__ prev,
                   const float* __restrict__ W_res,
                   const float* __restrict__ W_conv,
                   const float* __restrict__ in_norm,
                   const float* __restrict__ out_norm,
                   float* __restrict__ z, int n_tiles) {
    // Wp[kp][col] = {W[2kp][col], W[2kp+1][col]}, cols 0..63 = W_res, 64..127 = W_conv
    __shared__ float Wp[64 * SP];        // 73.7 KB of the 320 KB WGP LDS
    const int tid = threadIdx.x;
    for (int q = tid; q < 64 * 128; q += 256) {
        const int kp = q >> 7, col = q & 127;
        float w0, w1;
        if (col < OUT_DIM) {
            w0 = W_res[(2 * kp) * OUT_DIM + col];
            w1 = W_res[(2 * kp + 1) * OUT_DIM + col];
        } else {
            const int c = col - OUT_DIM;
            w0 = W_conv[(2 * kp) * OUT_DIM + c];
            w1 = W_conv[(2 * kp + 1) * OUT_DIM + c];
        }
        v2f wv; wv.x = w0; wv.y = w1;
        *(v2f*)&Wp[kp * SP + col * 2] = wv;
    }
    __syncthreads();

    const int wave  = tid >> 5;
    const int lane  = tid & 31;
    const int half  = lane >> 4;        // 0: lanes 0-15, 1: lanes 16-31
    const int m     = lane & 15;

    const int tile0 = blockIdx.x * 16 + wave;       // always < n_tiles (grid sized exactly)
    const int tile1 = tile0 + 8;                    // may be out of range in last block
    const bool t1ok = (tile1 < n_tiles);
    const int r0 = tile0 * 16;
    const int r1 = (t1ok ? tile1 : 0) * 16;         // harmless dummy rows when invalid

    // A (16x4) layout: lane holds prev[r+m][k + 2*half + {0,1}] -> one b64 load/step
    const float* __restrict__ ap0 = prev + (size_t)(r0 + m) * IN_DIM + 2 * half;
    const float* __restrict__ ap1 = prev + (size_t)(r1 + m) * IN_DIM + 2 * half;

    v8f acc0[8], acc1[8];
#pragma unroll
    for (int t = 0; t < 8; ++t) { acc0[t] = (v8f){}; acc1[t] = (v8f){}; }

#pragma unroll
    for (int kk = 0; kk < IN_DIM / 4; ++kk) {
        const v2f a0 = *(const v2f*)(ap0 + kk * 4);
        const v2f a1 = *(const v2f*)(ap1 + kk * 4);
        const float* __restrict__ bb = &Wp[(2 * kk + half) * SP + 2 * m];
#pragma unroll
        for (int t = 0; t < 8; ++t) {
            const v2f b = *(const v2f*)(bb + t * 32);   // ds_load_b64, conflict-free
            acc0[t] = __builtin_amdgcn_wmma_f32_16x16x4_f32(false, a0, false, b,
                                                            (short)0, acc0[t], false, false);
            acc1[t] = __builtin_amdgcn_wmma_f32_16x16x4_f32(false, a1, false, b,
                                                            (short)0, acc1[t], false, false);
        }
    }

    // C/D tile layout: VGPR g -> row (g + 8*half), col (m + 16*t)
#pragma unroll
    for (int g = 0; g < 8; ++g) {
        {
            const int rw = r0 + g + 8 * half;
            const float inr = in_norm[rw];
            const float onr = out_norm[rw];
            float* __restrict__ zr = z + (size_t)rw * OUT_DIM + m;
            zr[0]  = inr * acc0[0][g] + onr * acc0[4][g];
            zr[16] = inr * acc0[1][g] + onr * acc0[5][g];
            zr[32] = inr * acc0[2][g] + onr * acc0[6][g];
            zr[48] = inr * acc0[3][g] + onr * acc0[7][g];
        }
        if (t1ok) {
            const int rw = r1 + g + 8 * half;
            const float inr = in_norm[rw];
            const float onr = out_norm[rw];
            float* __restrict__ zr = z + (size_t)rw * OUT_DIM + m;
            zr[0]  = inr * acc1[0][g] + onr * acc1[4][g];
            zr[16] = inr * acc1[1][g] + onr * acc1[5][g];
            zr[32] = inr * acc1[2][g] + onr * acc1[6][g];
            zr[48] = inr * acc1[3][g] + onr * acc1[7][g];
        }
    }
}

// -------------------------------------------------------- fused edge scatter
// One wave32 per edge: gather z[src] (float2/lane, 256B coalesced),
// global_atomic_add_f32 into out[dst]. z and out are L2-resident (25.6 MB each
// in the 192 MB L2).
__global__ __launch_bounds__(256)
void scatter_kernel(const float* __restrict__ z,
                    const int* __restrict__ src, const int* __restrict__ dst,
                    float* __restrict__ out, int nE) {
    const int gwave = (blockIdx.x * blockDim.x + threadIdx.x) >> 5;
    const int lane  = threadIdx.x & 31;
    if (gwave >= nE) return;
    const int s = src[gwave];
    const int d = dst[gwave];
    v2f v = *(const v2f*)(z + (size_t)s * OUT_DIM + 2 * lane);
    float* o = out + (size_t)d * OUT_DIM + 2 * lane;
    atomicAdd(o,     v.x);
    atomicAdd(o + 1, v.y);
}

// ------------------------------------------------------------------ finalize
__global__ void finalize_kernel(float* __restrict__ out,
                                const float* __restrict__ in_norm,
                                const float* __restrict__ b_conv, int n) {
    int i = blockIdx.x * blockDim.x + threadIdx.x;
    if (i < n) {
        const int v = i >> 6;       // /OUT_DIM
        const int c = i & 63;
        float r = fmaf(in_norm[v], out[i], b_conv[c]);
        out[i] = fmaxf(r, 0.0f);
    }
}

// ---------------------------------------------------------------- dispatcher
extern "C" void kernel_launch(void* const* d_in, const int* in_sizes, int n_in,
                              void* d_out, int out_size, void* d_ws, size_t ws_size,
                              hipStream_t stream) {
    const float* prev   = (const float*)d_in[0];
    // d_in[1] ("raw") is unused by the reference
    const int*   src    = (const int*)d_in[2];
    const int*   dst    = (const int*)d_in[3];
    const float* W_res  = (const float*)d_in[4];
    const float* W_conv = (const float*)d_in[5];
    const float* b_conv = (const float*)d_in[6];
    float* out = (float*)d_out;

    // Workspace layout (floats): in_deg[N] | out_deg[N] | in_norm[N] | out_norm[N] | z[N*64]
    float* ws       = (float*)d_ws;
    float* in_deg   = ws;
    float* out_deg  = ws + (size_t)N_NODES;
    float* in_norm  = ws + (size_t)2 * N_NODES;
    float* out_norm = ws + (size_t)3 * N_NODES;
    float* z        = ws + (size_t)4 * N_NODES;

    const int n_out = N_NODES * OUT_DIM;

    // zero degree accumulators and the output accumulator (d_ws/d_out are poisoned)
    zero_kernel<<<(2 * N_NODES + 255) / 256, 256, 0, stream>>>(in_deg, 2 * N_NODES);
    zero_kernel<<<(n_out + 255) / 256, 256, 0, stream>>>(out, n_out);

    degree_kernel<<<(N_EDGES + 255) / 256, 256, 0, stream>>>(src, dst, out_deg, in_deg, N_EDGES);
    norm_kernel<<<(N_NODES + 255) / 256, 256, 0, stream>>>(in_deg, out_deg, in_norm, out_norm, N_NODES);

    // 16 tiles per block (8 waves x 2 tiles)
    gemm_z_kernel<<<(N_TILES + 15) / 16, 256, 0, stream>>>(prev, W_res, W_conv,
                                                           in_norm, out_norm, z, N_TILES);

    scatter_kernel<<<(N_EDGES + 7) / 8, 256, 0, stream>>>(z, src, dst, out, N_EDGES);

    finalize_kernel<<<(n_out + 255) / 256, 256, 0, stream>>>(out, in_norm, b_conv, n_out);
}